// PEG_Add_171798692025
// MI455X (gfx1250) — compile-verified
//
#include <hip/hip_runtime.h>
#include <hip/hip_bf16.h>
#include <stdint.h>

// Depthwise AdderNet 3x3 (stride 1, pad 1, groups=C) on x=(32,384,32,32) fp32.
//
// Roofline: ~50 MB read + ~50 MB write, ~0.34 GFLOP -> purely HBM-bound
// (~4.3 us at 23.3 TB/s). No contraction structure -> WMMA inapplicable;
// the gfx1250 path that matters is async global->LDS DMA (ASYNCcnt) to stage
// each contiguous 4 KB channel plane once, serving the 9x window reuse from
// LDS, with fully coalesced B128 traffic in both directions.
//
// 2 planes per 256-thread block: 2 async B128 DMAs in flight per thread
// before a single wait+barrier (more MLP, half the barriers per byte).
// Zero-fill is 3 unconditional ds_store_b128 per thread (tid masked to <256
// so the compiler can prove bounds and drop exec-mask guards).

#define LDS_STRIDE 40    // floats/row; interior at col 4 => 16B-aligned DMA writes
#define LDS_ROWS   34    // 32 interior rows + zero halo top/bottom
#define TILE_ALLOC 1536  // 34*40=1360 used; padded so 2*1536/4 = 768 = 3*256 float4
#define C_CH 384
#define HW   32

__global__ __launch_bounds__(256) void addernet_dw3x3_kernel(
    const float* __restrict__ x, const float* __restrict__ w,
    float* __restrict__ out)
{
    __shared__ __align__(16) float tile[2][TILE_ALLOC];   // 12 KB

    const int t      = threadIdx.x & 255;   // masked: compiler knows t < 256
    const int plane0 = blockIdx.x * 2;      // n*C + c, two consecutive planes
    const int c0     = plane0 % C_CH;
    const int c1     = (plane0 + 1) % C_CH;

    // 1) zero both tiles (halo zeros implement padding=1):
    //    flat 3072 floats = 768 float4 = 3 unconditional b128 stores/thread.
    float4* z = (float4*)&tile[0][0];
    const float4 zero4 = make_float4(0.f, 0.f, 0.f, 0.f);
    #pragma unroll
    for (int k = 0; k < 3; ++k) z[t + 256 * k] = zero4;
    __syncthreads();

    // 2) async DMA both 32x32 planes into LDS interiors.
    //    Each thread moves one 16B chunk per plane: 256*16B = whole plane.
    const int row = t >> 3;          // 0..31
    const int c4  = (t & 7) << 2;    // 0,4,...,28
    const float* gp0 = x + ((size_t)plane0 << 10) + (size_t)(row * HW + c4);
    const float* gp1 = gp0 + (HW * HW);
    const int    ldsIdx = (row + 1) * LDS_STRIDE + 4 + c4;
    unsigned lds0 = (unsigned)(uintptr_t)(&tile[0][ldsIdx]);
    unsigned lds1 = (unsigned)(uintptr_t)(&tile[1][ldsIdx]);
    asm volatile("global_load_async_to_lds_b128 %0, %1, off"
                 :: "v"(lds0), "v"(gp0) : "memory");
    asm volatile("global_load_async_to_lds_b128 %0, %1, off"
                 :: "v"(lds1), "v"(gp1) : "memory");
    asm volatile("s_wait_asynccnt 0x0" ::: "memory");
    __syncthreads();

    // 3) per-channel 3x3 weights: workgroup-uniform address -> scalar loads
    float wr0[9], wr1[9];
    #pragma unroll
    for (int i = 0; i < 9; ++i) wr0[i] = w[c0 * 9 + i];
    #pragma unroll
    for (int i = 0; i < 9; ++i) wr1[i] = w[c1 * 9 + i];

    // 4) each thread computes 4 consecutive outputs per plane from a 3x6
    //    LDS window (consecutive lane addresses -> conflict-free).
    const size_t obase = ((size_t)plane0 << 10) + (size_t)(row * HW + c4);

    #pragma unroll
    for (int p = 0; p < 2; ++p) {
        const float* wr = (p == 0) ? wr0 : wr1;
        float win[3][6];
        #pragma unroll
        for (int u = 0; u < 3; ++u) {
            #pragma unroll
            for (int m = 0; m < 6; ++m)
                win[u][m] = tile[p][(row + u) * LDS_STRIDE + (c4 + 3 + m)];
        }
        float acc0 = 0.f, acc1 = 0.f, acc2 = 0.f, acc3 = 0.f;
        #pragma unroll
        for (int u = 0; u < 3; ++u) {
            #pragma unroll
            for (int v = 0; v < 3; ++v) {
                const float wv = wr[u * 3 + v];
                acc0 -= fabsf(win[u][v + 0] - wv);
                acc1 -= fabsf(win[u][v + 1] - wv);
                acc2 -= fabsf(win[u][v + 2] - wv);
                acc3 -= fabsf(win[u][v + 3] - wv);
            }
        }
        // 5) coalesced float4 store (global_store_b128)
        float4 r = make_float4(acc0, acc1, acc2, acc3);
        *(float4*)(out + obase + (size_t)(p * HW * HW)) = r;
    }
}

extern "C" void kernel_launch(void* const* d_in, const int* in_sizes, int n_in,
                              void* d_out, int out_size, void* d_ws, size_t ws_size,
                              hipStream_t stream) {
    const float* x  = (const float*)d_in[0];   // (32,384,32,32) fp32
    const float* w  = (const float*)d_in[1];   // (384,3,3) fp32
    float* out      = (float*)d_out;           // (32,384,32,32) fp32
    const int planes = in_sizes[0] / (HW * HW);   // B*C = 12288 (even)
    addernet_dw3x3_kernel<<<planes / 2, 256, 0, stream>>>(x, w, out);
}